// DeepWalk_34462817583811
// MI455X (gfx1250) — compile-verified
//
#include <hip/hip_runtime.h>

typedef __attribute__((ext_vector_type(2))) float v2f;
typedef __attribute__((ext_vector_type(8))) float v8f;

#define EMB 128
#define PAIRS_PER_WAVE 16
#define WAVES_PER_BLOCK 8
#define PAIRS_PER_BLOCK (PAIRS_PER_WAVE * WAVES_PER_BLOCK)

// Each wave32 handles 16 (center, context) pairs.
// A = 16x4 tile of center embeddings (rows = pairs), B = 4x16 tile of context
// embeddings (cols = pairs). Chain V_WMMA_F32_16X16X4_F32 over K=128, then
// extract the 16x16 result's diagonal = per-pair dot products.
__global__ __launch_bounds__(256) void deepwalk_sgns_wmma(
    const float* __restrict__ phi,      // [N_NODES, 128]
    const float* __restrict__ phi_out,  // [N_NODES, 128]
    const int*   __restrict__ centers,  // [P] int32
    const int*   __restrict__ contexts, // [P] int32
    float*       __restrict__ out,      // [P] loss
    int nPairs)
{
    const int lane  = threadIdx.x & 31;
    const int wave  = threadIdx.x >> 5;
    const int pairBase = (blockIdx.x * WAVES_PER_BLOCK + wave) * PAIRS_PER_WAVE;

    // WMMA f32 16x4 A layout: lane l feeds row M = l%16 with K = 2*(l>>4)+{0,1}.
    // B 4x16 mirrors with N = l%16 -> identical per-lane addressing on phi_out.
    const int m     = lane & 15;   // pair slot this lane feeds
    const int khalf = lane >> 4;   // K sub-offset: 0 -> K%4 in {0,1}, 1 -> {2,3}

    int p  = pairBase + m;
    int pc = (p < nPairs) ? p : 0;            // clamp (select, no divergence)
    const long long crow = (long long)centers[pc];
    const long long krow = (long long)contexts[pc];
    const float* arow = phi     + crow * EMB + 2 * khalf;
    const float* brow = phi_out + krow * EMB + 2 * khalf;

    v8f acc0 = {};
    v8f acc1 = {};
    #pragma unroll
    for (int i = 0; i < EMB / 8; ++i) {
        v2f a0 = *(const v2f*)(arow + 8 * i);       // K chunk 8i..8i+3
        v2f b0 = *(const v2f*)(brow + 8 * i);
        v2f a1 = *(const v2f*)(arow + 8 * i + 4);   // K chunk 8i+4..8i+7
        v2f b1 = *(const v2f*)(brow + 8 * i + 4);
        acc0 = __builtin_amdgcn_wmma_f32_16x16x4_f32(
            false, a0, false, b0, (short)0, acc0, false, false);
        acc1 = __builtin_amdgcn_wmma_f32_16x16x4_f32(
            false, a1, false, b1, (short)0, acc1, false, false);
    }
    v8f acc = acc0 + acc1;

    // Diagonal extraction (C/D 16x16 f32 layout):
    //   m in 0..7  : lane m,     VGPR m
    //   m in 8..15 : lane m+16,  VGPR m-8   (i.e. lanes 24..31, VGPR lane-24)
    const bool valid = (lane < 8) | (lane >= 24);
    const int  idx   = (lane < 8) ? lane : (lane - 24);  // VGPR index 0..7
    float d = acc[0];
    d = (idx == 1) ? acc[1] : d;
    d = (idx == 2) ? acc[2] : d;
    d = (idx == 3) ? acc[3] : d;
    d = (idx == 4) ? acc[4] : d;
    d = (idx == 5) ? acc[5] : d;
    d = (idx == 6) ? acc[6] : d;
    d = (idx == 7) ? acc[7] : d;

    const int outIdx = pairBase + ((lane < 8) ? lane : (lane - 16));
    if (valid && outIdx < nPairs) {
        // loss = softplus(-x) = max(-x, 0) + log1p(exp(-|x|))  (stable)
        float x = d;
        float loss = fmaxf(-x, 0.0f) + log1pf(__expf(-fabsf(x)));
        out[outIdx] = loss;
    }
}

extern "C" void kernel_launch(void* const* d_in, const int* in_sizes, int n_in,
                              void* d_out, int out_size, void* d_ws, size_t ws_size,
                              hipStream_t stream) {
    const float* phi      = (const float*)d_in[0];
    const float* phi_out  = (const float*)d_in[1];
    const int*   centers  = (const int*)d_in[2];
    const int*   contexts = (const int*)d_in[3];
    float*       out      = (float*)d_out;

    const int nPairs = in_sizes[2];
    const int blocks = (nPairs + PAIRS_PER_BLOCK - 1) / PAIRS_PER_BLOCK;
    deepwalk_sgns_wmma<<<blocks, 256, 0, stream>>>(phi, phi_out, centers,
                                                   contexts, out, nPairs);
}